// DiscGCN_80934363726533
// MI455X (gfx1250) — compile-verified
//
#include <hip/hip_runtime.h>
#include <cstdint>

typedef float v2f __attribute__((ext_vector_type(2)));
typedef float v8f __attribute__((ext_vector_type(8)));

#define N_NODES 100000
#define N_EDGES 1600000
#define DIM     128

// Guaranteed native no-return fp32 atomic add (STOREcnt path, no CAS loop).
__device__ __forceinline__ void atomic_fadd_f32(float* p, float v) {
    asm volatile("global_atomic_add_f32 %0, %1, off"
                 :
                 : "v"(p), "v"(v)
                 : "memory");
}

// ---------------------------------------------------------------------------
// Degree / normalization
// ---------------------------------------------------------------------------
__global__ void k_count_deg(const int* __restrict__ dst, uint32_t* __restrict__ cnt, int E) {
    int i  = blockIdx.x * blockDim.x + threadIdx.x;
    int st = gridDim.x * blockDim.x;
    for (; i < E; i += st) {
        __hip_atomic_fetch_add(&cnt[dst[i]], 1u, __ATOMIC_RELAXED, __HIP_MEMORY_SCOPE_AGENT);
    }
}

__global__ void k_dinv(const uint32_t* __restrict__ cnt, float* __restrict__ dinv, int n) {
    int i = blockIdx.x * blockDim.x + threadIdx.x;
    if (i < n) dinv[i] = rsqrtf(1.0f + (float)cnt[i]);
}

// ---------------------------------------------------------------------------
// fp32 WMMA GEMM: Out[M,128] = A[M,128] @ W[128,128]
// block = 256 threads = 8 waves; blockIdx.x = M-tile (16 rows),
// wave id = N-tile (16 cols). Uses V_WMMA_F32_16X16X4_F32, K stepped by 4.
// ---------------------------------------------------------------------------
__global__ void k_gemm128(const float* __restrict__ A, const float* __restrict__ W,
                          float* __restrict__ Out, int M) {
    const int lane  = threadIdx.x & 31;
    const int wave  = threadIdx.x >> 5;     // 0..7 -> N tile
    const int m0    = blockIdx.x * 16;
    const int n0    = wave * 16;
    if (m0 >= M) return;

    const int mrow  = m0 + (lane & 15);     // A-frag row for this lane
    const int ncol  = n0 + (lane & 15);     // B-frag col for this lane
    const int khalf = (lane >> 4) << 1;     // lanes 16-31 hold K+2,K+3

    v8f c = {};
    const float* arow = A + (size_t)mrow * DIM;
    #pragma unroll
    for (int k = 0; k < DIM; k += 4) {
        const int kb = k + khalf;
        v2f a, b;
        a.x = arow[kb];
        a.y = arow[kb + 1];
        b.x = W[(size_t)kb * DIM + ncol];
        b.y = W[(size_t)(kb + 1) * DIM + ncol];
        c = __builtin_amdgcn_wmma_f32_16x16x4_f32(
                /*neg_a=*/false, a, /*neg_b=*/false, b,
                /*c_mod=*/(short)0, c, /*reuse_a=*/false, /*reuse_b=*/false);
    }
    // C/D layout: VGPR v -> M = v + 8*(lane>>4), N = lane&15
    const int mbase = m0 + 8 * (lane >> 4);
    #pragma unroll
    for (int v = 0; v < 8; ++v) {
        Out[(size_t)(mbase + v) * DIM + ncol] = c[v];
    }
}

// ---------------------------------------------------------------------------
// Edge aggregation, 128-dim: one wave per edge, float4 per lane.
// agg[dst] += dinv[src]*dinv[dst] * h[src]   (native f32 atomics, no return)
// ---------------------------------------------------------------------------
__global__ void k_agg128(const float* __restrict__ h, const int* __restrict__ src,
                         const int* __restrict__ dst, const float* __restrict__ dinv,
                         float* __restrict__ agg, int E) {
    const int lane = threadIdx.x & 31;
    int w  = (blockIdx.x * blockDim.x + threadIdx.x) >> 5;
    const int nw = (gridDim.x * blockDim.x) >> 5;
    for (; w < E; w += nw) {
        const int s = src[w];
        const int d = dst[w];
        const float nrm = dinv[s] * dinv[d];
        const float4 hv = ((const float4*)(h + (size_t)s * DIM))[lane];
        float* base = agg + (size_t)d * DIM + lane * 4;
        atomic_fadd_f32(base + 0, nrm * hv.x);
        atomic_fadd_f32(base + 1, nrm * hv.y);
        atomic_fadd_f32(base + 2, nrm * hv.z);
        atomic_fadd_f32(base + 3, nrm * hv.w);
    }
}

// ---------------------------------------------------------------------------
// Epilogue: agg <- relu(agg + dinv[node]^2 * h + b[col])   (in place)
// ---------------------------------------------------------------------------
__global__ void k_fin_relu(float* __restrict__ agg, const float* __restrict__ h,
                           const float* __restrict__ dinv, const float* __restrict__ b,
                           int total) {
    int i = blockIdx.x * blockDim.x + threadIdx.x;
    if (i >= total) return;
    const int node = i >> 7;
    const int col  = i & 127;
    const float di = dinv[node];
    const float v  = agg[i] + di * di * h[i] + b[col];
    agg[i] = fmaxf(v, 0.0f);
}

// ---------------------------------------------------------------------------
// GEMV for W3 [128,1]: one wave per node, butterfly reduce.
// ---------------------------------------------------------------------------
__global__ void k_gemv_w3(const float* __restrict__ z, const float* __restrict__ w3,
                          float* __restrict__ h3, int n) {
    const int lane = threadIdx.x & 31;
    const int node = (blockIdx.x * blockDim.x + threadIdx.x) >> 5;
    if (node >= n) return;
    const float4 zv = ((const float4*)(z + (size_t)node * DIM))[lane];
    const float4 wv = ((const float4*)w3)[lane];
    float p = zv.x * wv.x + zv.y * wv.y + zv.z * wv.z + zv.w * wv.w;
    #pragma unroll
    for (int m = 16; m >= 1; m >>= 1) p += __shfl_xor(p, m, 32);
    if (lane == 0) h3[node] = p;
}

// ---------------------------------------------------------------------------
// Scalar edge aggregation for layer 3: agg3[dst] += norm * h3[src]
// ---------------------------------------------------------------------------
__global__ void k_agg1(const float* __restrict__ h3, const int* __restrict__ src,
                       const int* __restrict__ dst, const float* __restrict__ dinv,
                       float* __restrict__ agg3, int E) {
    int i  = blockIdx.x * blockDim.x + threadIdx.x;
    int st = gridDim.x * blockDim.x;
    for (; i < E; i += st) {
        const int s = src[i];
        const int d = dst[i];
        atomic_fadd_f32(&agg3[d], dinv[s] * dinv[d] * h3[s]);
    }
}

__global__ void k_fin_sig(const float* __restrict__ agg3, const float* __restrict__ h3,
                          const float* __restrict__ dinv, const float* __restrict__ b3,
                          float* __restrict__ out, int n) {
    int i = blockIdx.x * blockDim.x + threadIdx.x;
    if (i >= n) return;
    const float di = dinv[i];
    const float v  = agg3[i] + di * di * h3[i] + b3[0];
    out[i] = 1.0f / (1.0f + __expf(-v));
}

// ---------------------------------------------------------------------------
// Launch
// ---------------------------------------------------------------------------
extern "C" void kernel_launch(void* const* d_in, const int* in_sizes, int n_in,
                              void* d_out, int out_size, void* d_ws, size_t ws_size,
                              hipStream_t stream) {
    (void)in_sizes; (void)n_in; (void)out_size; (void)ws_size;
    const float* x  = (const float*)d_in[0];
    const int*   ei = (const int*)d_in[1];       // [2, E] int
    const float* W1 = (const float*)d_in[2];
    const float* b1 = (const float*)d_in[3];
    const float* W2 = (const float*)d_in[4];
    const float* b2 = (const float*)d_in[5];
    const float* W3 = (const float*)d_in[6];
    const float* b3 = (const float*)d_in[7];
    float* out = (float*)d_out;

    const int* src = ei;            // ei[0, :]
    const int* dst = ei + N_EDGES;  // ei[1, :]

    // Workspace layout (bytes)
    char* ws = (char*)d_ws;
    const size_t featBytes = (size_t)N_NODES * DIM * sizeof(float);   // 51.2 MB
    uint32_t* cnt  = (uint32_t*)(ws);                       // 400 KB
    float*    dinv = (float*)(ws + (512ull << 10));         // 400 KB
    float*    h3   = (float*)(ws + (1024ull << 10));        // 400 KB
    float*    agg3 = (float*)(ws + (1536ull << 10));        // 400 KB
    float*    bufA = (float*)(ws + (2048ull << 10));        // h1 / h2
    float*    bufB = (float*)((char*)bufA + featBytes);     // agg/z ping-pong

    const int total = N_NODES * DIM;

    // --- normalization (graph-constant, computed in-launch for determinism)
    hipMemsetAsync(cnt, 0, (size_t)N_NODES * sizeof(uint32_t), stream);
    k_count_deg<<<2048, 256, 0, stream>>>(dst, cnt, N_EDGES);
    k_dinv<<<(N_NODES + 255) / 256, 256, 0, stream>>>(cnt, dinv, N_NODES);

    // --- layer 1: h1 = x@W1 (bufA); agg (bufB); z1 = relu(...) in bufB
    k_gemm128<<<N_NODES / 16, 256, 0, stream>>>(x, W1, bufA, N_NODES);
    hipMemsetAsync(bufB, 0, featBytes, stream);
    k_agg128<<<8192, 256, 0, stream>>>(bufA, src, dst, dinv, bufB, N_EDGES);
    k_fin_relu<<<(total + 255) / 256, 256, 0, stream>>>(bufB, bufA, dinv, b1, total);

    // --- layer 2: h2 = z1@W2 (bufA); reuse bufB as agg; z2 in bufB
    k_gemm128<<<N_NODES / 16, 256, 0, stream>>>(bufB, W2, bufA, N_NODES);
    hipMemsetAsync(bufB, 0, featBytes, stream);
    k_agg128<<<8192, 256, 0, stream>>>(bufA, src, dst, dinv, bufB, N_EDGES);
    k_fin_relu<<<(total + 255) / 256, 256, 0, stream>>>(bufB, bufA, dinv, b2, total);

    // --- layer 3: h3 = z2@W3; scalar aggregate; sigmoid
    k_gemv_w3<<<(N_NODES * 32 + 255) / 256, 256, 0, stream>>>(bufB, W3, h3, N_NODES);
    hipMemsetAsync(agg3, 0, (size_t)N_NODES * sizeof(float), stream);
    k_agg1<<<2048, 256, 0, stream>>>(h3, src, dst, dinv, agg3, N_EDGES);
    k_fin_sig<<<(N_NODES + 255) / 256, 256, 0, stream>>>(agg3, h3, dinv, b3, out, N_NODES);
}